// OrthogonalMatrix_37589553774838
// MI455X (gfx1250) — compile-verified
//
#include <hip/hip_runtime.h>
#include <hip/hip_bf16.h>
#include <math.h>
#include <stdint.h>

#define DIM 2048
#define PB  32          // panel width

typedef __attribute__((ext_vector_type(2))) float v2f;
typedef __attribute__((ext_vector_type(8))) float v8f;
typedef unsigned int u32x4 __attribute__((ext_vector_type(4)));
typedef int i32x4 __attribute__((ext_vector_type(4)));
typedef int i32x8 __attribute__((ext_vector_type(8)));

// ---------------------------------------------------------------------------
// Tensor Data Mover: async 2D tile load Global -> LDS.
// Descriptor per CDNA5 ISA ch.8 (D# group0/group1, 2D tensor, 4-byte elems).
// Issued by one wave; completion via s_wait_tensorcnt.
// ---------------------------------------------------------------------------
__device__ __forceinline__ void tdm_load_2d(const void* gptr, unsigned ldsOff,
                                            unsigned tensor_w, unsigned tensor_h,
                                            unsigned tile_w, unsigned tile_h,
                                            unsigned row_stride_elems) {
    unsigned long long ga = (unsigned long long)(uintptr_t)gptr;
    u32x4 g0;
    g0[0] = 1u;                                     // count=1, user descriptor
    g0[1] = ldsOff;                                 // LDS byte address
    g0[2] = (unsigned)ga;                           // global_addr[31:0]
    g0[3] = (unsigned)((ga >> 32) & 0x01FFFFFFu)    // global_addr[56:32]
          | (2u << 30);                             // type = 2 ("image")
    i32x8 g1;
    g1[0] = (int)(2u << 16);                        // data_size = 2 (4 bytes)
    g1[1] = (int)((tensor_w & 0xFFFFu) << 16);      // tensor_dim0 lo16
    g1[2] = (int)((tensor_w >> 16) | ((tensor_h & 0xFFFFu) << 16));
    g1[3] = (int)((tensor_h >> 16) | (tile_w << 16));   // tile_dim0
    g1[4] = (int)(tile_h & 0xFFFFu);                    // tile_dim1 (tile_dim2=0)
    g1[5] = (int)row_stride_elems;                      // tensor_dim0_stride lo32
    g1[6] = 0;
    g1[7] = 0;
    i32x4 z4 = {0, 0, 0, 0};
#if defined(__clang_major__) && (__clang_major__ >= 23)
    i32x8 z8 = {0, 0, 0, 0, 0, 0, 0, 0};
    __builtin_amdgcn_tensor_load_to_lds(g0, g1, z4, z4, z8, 0);
#else
    __builtin_amdgcn_tensor_load_to_lds(g0, g1, z4, z4, 0);
#endif
}

// ---------------------------------------------------------------------------
// Panel factorization: sequential MGS on columns [p0, p0+PB) of X (row-major
// 2048x2048). One workgroup of 1024 threads; thread t owns rows t and t+1024.
// Panel register-resident (fully unrolled -> constant indices, no scratch).
// ---------------------------------------------------------------------------
__global__ void __launch_bounds__(1024)
mgs_panel(float* __restrict__ X, int p0) {
    __shared__ float wsum[32][PB];
    __shared__ float red[PB];

    const int tid  = threadIdx.x;
    const int lane = tid & 31;
    const int wid  = tid >> 5;
    const int r0 = tid;
    const int r1 = tid + 1024;

    float x0[PB], x1[PB];
#pragma unroll
    for (int j = 0; j < PB; ++j) {
        x0[j] = X[(size_t)r0 * DIM + p0 + j];
        x1[j] = X[(size_t)r1 * DIM + p0 + j];
    }

#pragma unroll
    for (int i = 0; i < PB; ++i) {
        // ---- norm of column i ----
        float part = x0[i] * x0[i] + x1[i] * x1[i];
#pragma unroll
        for (int off = 16; off > 0; off >>= 1)
            part += __shfl_xor(part, off, 32);
        if (lane == 0) wsum[wid][0] = part;
        __syncthreads();
        if (tid == 0) {
            float s = 0.0f;
#pragma unroll
            for (int ww = 0; ww < 32; ++ww) s += wsum[ww][0];
            red[0] = 1.0f / sqrtf(s);
        }
        __syncthreads();
        const float inv = red[0];
        x0[i] *= inv;
        x1[i] *= inv;
        // no barrier needed: following writes touch wsum[*][j>=1]/red[j>=1] only

        // ---- dots q_i . x_j for j > i ----
#pragma unroll
        for (int j = i + 1; j < PB; ++j) {
            float p = x0[i] * x0[j] + x1[i] * x1[j];
#pragma unroll
            for (int off = 16; off > 0; off >>= 1)
                p += __shfl_xor(p, off, 32);
            if (lane == 0) wsum[wid][j] = p;
        }
        __syncthreads();
        if (tid < PB - i - 1) {
            const int j = i + 1 + tid;
            float s = 0.0f;
#pragma unroll
            for (int ww = 0; ww < 32; ++ww) s += wsum[ww][j];
            red[j] = s;
        }
        __syncthreads();

        // ---- rank-1 update of remaining panel columns ----
#pragma unroll
        for (int j = i + 1; j < PB; ++j) {
            const float pj = red[j];
            x0[j] -= pj * x0[i];
            x1[j] -= pj * x1[i];
        }
        __syncthreads();
    }

#pragma unroll
    for (int j = 0; j < PB; ++j) {
        X[(size_t)r0 * DIM + p0 + j] = x0[j];
        X[(size_t)r1 * DIM + p0 + j] = x1[j];
    }
}

// ---------------------------------------------------------------------------
// S = Q_panel^T * X_trail : S[i][j] = sum_k X[k][p0+i] * X[k][c0+j]
// LDS-tiled, software-pipelined GEMM: block = 128 threads (4 waves), block
// tile 32(M) x 64(N), K-tile 16. K-tile k+1 is fetched to registers while
// tile k is consumed from LDS. Pitch-18 K-innermost layout -> conflict-free
// ds_load_b64 fragment reads. Each wave: 2 accumulators (A fragment reuse).
// ---------------------------------------------------------------------------
#define QTX_KT 16
__global__ void __launch_bounds__(128)
mgs_qtx(const float* __restrict__ X, float* __restrict__ S,
        int p0, int c0, int T) {
    __shared__ float ldsA[32 * 18];   // [panelCol][k], pitch 18
    __shared__ float ldsB[64 * 18];   // [trailCol][k], pitch 18

    const int tid  = threadIdx.x;
    const int lane = tid & 31;
    const int w    = tid >> 5;
    const int m    = lane & 15;
    const int hi   = lane >> 4;
    const int mi   = w & 1;                 // M tile (0..1)
    const int nq   = w >> 1;                // N half (0..1)
    const int cb0  = c0 + blockIdx.x * 64;  // first trailing col of block

    // staging coordinates
    const int ac = tid & 31;                // panel col
    const int ar = tid >> 5;                // k row (0..3, step 4)
    const int bc = tid & 63;                // trailing col
    const int br = tid >> 6;                // k row (0..1, step 2)
    int bcol = cb0 + bc;
    if (bcol > DIM - 1) bcol = DIM - 1;     // clamp ragged N (stores guarded)

    float aReg[4], bReg[8];

    // prologue: fetch K-tile 0 and commit to LDS
#pragma unroll
    for (int rr = 0; rr < 4; ++rr)
        aReg[rr] = X[(size_t)(ar + rr * 4) * DIM + p0 + ac];
#pragma unroll
    for (int rr = 0; rr < 8; ++rr)
        bReg[rr] = X[(size_t)(br + rr * 2) * DIM + bcol];
#pragma unroll
    for (int rr = 0; rr < 4; ++rr) ldsA[ac * 18 + ar + rr * 4] = aReg[rr];
#pragma unroll
    for (int rr = 0; rr < 8; ++rr) ldsB[bc * 18 + br + rr * 2] = bReg[rr];
    __syncthreads();

    v8f acc0 = {}, acc1 = {};

    for (int kk = QTX_KT; kk <= DIM; kk += QTX_KT) {
        // fetch next K-tile into registers (overlaps with compute below)
        if (kk < DIM) {
#pragma unroll
            for (int rr = 0; rr < 4; ++rr)
                aReg[rr] = X[(size_t)(kk + ar + rr * 4) * DIM + p0 + ac];
#pragma unroll
            for (int rr = 0; rr < 8; ++rr)
                bReg[rr] = X[(size_t)(kk + br + rr * 2) * DIM + bcol];
        }

        // compute current K-tile from LDS
#pragma unroll
        for (int kb = 0; kb < QTX_KT; kb += 4) {
            const int kbl = kb + 2 * hi;
            v2f a, b0, b1;
            a.x  = ldsA[(mi * 16 + m) * 18 + kbl];
            a.y  = ldsA[(mi * 16 + m) * 18 + kbl + 1];
            b0.x = ldsB[(nq * 32 + m) * 18 + kbl];
            b0.y = ldsB[(nq * 32 + m) * 18 + kbl + 1];
            b1.x = ldsB[(nq * 32 + 16 + m) * 18 + kbl];
            b1.y = ldsB[(nq * 32 + 16 + m) * 18 + kbl + 1];
            acc0 = __builtin_amdgcn_wmma_f32_16x16x4_f32(
                false, a, false, b0, (short)0, acc0, false, false);
            acc1 = __builtin_amdgcn_wmma_f32_16x16x4_f32(
                false, a, false, b1, (short)0, acc1, false, false);
        }

        if (kk < DIM) {
            __syncthreads();   // all waves done reading LDS
#pragma unroll
            for (int rr = 0; rr < 4; ++rr) ldsA[ac * 18 + ar + rr * 4] = aReg[rr];
#pragma unroll
            for (int rr = 0; rr < 8; ++rr) ldsB[bc * 18 + br + rr * 2] = bReg[rr];
            __syncthreads();   // LDS tile ready
        }
    }

    // store S tile; validity is wave-uniform (base and T are multiples of 32,
    // so base < T covers both 16-wide N tiles of this wave)
    const int jbase = blockIdx.x * 64 + nq * 32;
    if (jbase < T) {
#pragma unroll
        for (int v = 0; v < 8; ++v) {
            const int row = mi * 16 + v + hi * 8;
            S[(size_t)row * T + jbase + m]      = acc0[v];
            S[(size_t)row * T + jbase + 16 + m] = acc1[v];
        }
    }
}

// ---------------------------------------------------------------------------
// X_trail -= Q_panel * S : block = 256 threads (8 waves), block tile
// 64(M) x 64(N), K = PB = 32 fully unrolled (16 WMMAs/wave).
// B operand (S tile, 32 x width) staged Global->LDS by the Tensor Data Mover
// (wave 0), overlapped with coalesced A-tile staging by all threads.
// ---------------------------------------------------------------------------
__global__ void __launch_bounds__(256)
mgs_update(float* __restrict__ X, const float* __restrict__ S,
           int p0, int c0, int T) {
    __shared__ float ldsA[64 * 34];   // [row][k], pitch 34
    __shared__ float ldsS[32 * 64];   // [k][col], pitch = width (TDM packs rows)

    const int tid  = threadIdx.x;
    const int lane = tid & 31;
    const int w    = tid >> 5;
    const int m    = lane & 15;
    const int hi   = lane >> 4;
    const int mi   = w & 3;                 // M tile (0..3)
    const int nq   = w >> 2;                // N half (0..1)
    const int jb0  = blockIdx.x * 64;       // col offset inside trailing block
    const int r0   = blockIdx.y * 64;       // row offset

    int width = T - jb0;
    if (width > 64) width = 64;

    // Async TDM: S rows [0,32) x cols [jb0, jb0+width) -> ldsS (pitch = width)
    if (w == 0) {
        tdm_load_2d(S + jb0, (unsigned)(uintptr_t)ldsS,
                    (unsigned)T, 32u, (unsigned)width, 32u, (unsigned)T);
    }

    // Stage A: X rows [r0, r0+64) x panel cols [p0, p0+32) -> ldsA
    {
        const int ac = tid & 31;
        const int ar = tid >> 5;
#pragma unroll
        for (int rr = 0; rr < 64; rr += 8) {
            const int r = ar + rr;
            ldsA[r * 34 + ac] = X[(size_t)(r0 + r) * DIM + p0 + ac];
        }
    }
    if (w == 0) __builtin_amdgcn_s_wait_tensorcnt(0);
    __syncthreads();

    v8f acc0 = {}, acc1 = {};
#pragma unroll
    for (int k = 0; k < PB; k += 4) {
        const int kbl = k + 2 * hi;
        v2f a, b0, b1;
        a.x = ldsA[(mi * 16 + m) * 34 + kbl];
        a.y = ldsA[(mi * 16 + m) * 34 + kbl + 1];
        // unguarded LDS reads: ragged tiles stay inside the 32*64 array and
        // their results are never stored; keeps EXEC full and VALU clean
        const int n0 = nq * 32 + m;
        const int n1 = n0 + 16;
        b0.x = ldsS[(kbl)     * width + n0];
        b0.y = ldsS[(kbl + 1) * width + n0];
        b1.x = ldsS[(kbl)     * width + n1];
        b1.y = ldsS[(kbl + 1) * width + n1];
        acc0 = __builtin_amdgcn_wmma_f32_16x16x4_f32(
            false, a, false, b0, (short)0, acc0, false, false);
        acc1 = __builtin_amdgcn_wmma_f32_16x16x4_f32(
            false, a, false, b1, (short)0, acc1, false, false);
    }

    // RMW epilogue; wave-uniform validity (multiples of 32)
    const int jbase = jb0 + nq * 32;
    if (jbase < T) {
#pragma unroll
        for (int v = 0; v < 8; ++v) {
            const int row = r0 + mi * 16 + v + hi * 8;
            X[(size_t)row * DIM + c0 + jbase + m]      -= acc0[v];
            X[(size_t)row * DIM + c0 + jbase + 16 + m] -= acc1[v];
        }
    }
}

// ---------------------------------------------------------------------------
// Host launcher: copy input -> output, then blocked right-looking MGS.
// All launches on `stream`; workspace S in d_ws (<= 32*2016*4 = 258 KB).
// ---------------------------------------------------------------------------
extern "C" void kernel_launch(void* const* d_in, const int* in_sizes, int n_in,
                              void* d_out, int out_size, void* d_ws, size_t ws_size,
                              hipStream_t stream) {
    (void)in_sizes; (void)n_in; (void)out_size; (void)ws_size;
    const float* W = (const float*)d_in[0];
    float* X = (float*)d_out;     // factor in place in the output buffer
    float* S = (float*)d_ws;      // PB x T workspace

    hipMemcpyAsync(X, W, (size_t)DIM * DIM * sizeof(float),
                   hipMemcpyDeviceToDevice, stream);

    for (int p = 0; p < DIM / PB; ++p) {
        const int p0 = p * PB;
        mgs_panel<<<1, 1024, 0, stream>>>(X, p0);

        const int c0 = p0 + PB;
        const int T  = DIM - c0;
        if (T > 0) {
            dim3 gq((T + 63) / 64, 1);
            mgs_qtx<<<gq, 128, 0, stream>>>(X, S, p0, c0, T);
            dim3 gu((T + 63) / 64, DIM / 64);
            mgs_update<<<gu, 256, 0, stream>>>(X, S, p0, c0, T);
        }
    }
}